// AdaptiveHyperbolicTripletLoss_16887811408100
// MI455X (gfx1250) — compile-verified
//
#include <hip/hip_runtime.h>
#include <cstdint>
#include <math.h>

// ---------------------------------------------------------------------------
// AdaptiveHyperbolicTripletLoss for MI455X (gfx1250, wave32)
//
//   1) build_lists   : parallel counting-sort -> per-class index lists + ranks
//   2) norms_kernel  : ||x||^2 for all rows (one row per wave)
//   3) sample_kernel : threefry2x32 (JAX-compatible), constexpr split keys
//   4) triplet_kernel: anchor tile staged via ASYNC global->LDS (b128) when
//                      available; f32 WMMA 16x16x4 with clause-grouped B
//                      loads; diag extraction by shuffle; hyperbolic loss
//   5) finalize      : loss / num_active / total / active_ratio
// ---------------------------------------------------------------------------

#define B_SZ   8192
#define DIM    128
#define K_TRIP 5
#define NCLS   64
#define EPSF   1e-7f

// workspace layout (4-byte units)
#define WS_CNT    0
#define WS_OFF    (WS_CNT + NCLS)
#define WS_LIST   (WS_OFF + NCLS)
#define WS_RANK   (WS_LIST + B_SZ)
#define WS_POSIDX (WS_RANK + B_SZ)
#define WS_NEGIDX (WS_POSIDX + B_SZ*K_TRIP)
#define WS_NORM   (WS_NEGIDX + B_SZ*K_TRIP)
#define WS_ACC    (WS_NORM + B_SZ)          // [0]=sum(f32) [1]=active(i32) [2]=valid(i32)

typedef float v2f __attribute__((ext_vector_type(2)));
typedef float v8f __attribute__((ext_vector_type(8)));
typedef int   v4i __attribute__((ext_vector_type(4)));

#if __has_builtin(__builtin_amdgcn_global_load_async_to_lds_b128) && \
    __has_builtin(__builtin_amdgcn_s_wait_asynccnt)
#define USE_ASYNC_LDS 1
#define AS_GLOBAL __attribute__((address_space(1)))
#define AS_LDS    __attribute__((address_space(3)))
#else
#define USE_ASYNC_LDS 0
#endif

// ------------------------------- threefry ----------------------------------
__host__ __device__ constexpr uint32_t rotl32(uint32_t x, int n) {
  return (x << n) | (x >> (32 - n));
}

struct tf2 { uint32_t x0, x1; };

__host__ __device__ constexpr tf2 threefry2x32(uint32_t k0, uint32_t k1,
                                               uint32_t c0, uint32_t c1) {
  const uint32_t ks0 = k0, ks1 = k1, ks2 = k0 ^ k1 ^ 0x1BD11BDAu;
  uint32_t x0 = c0 + ks0, x1 = c1 + ks1;
  const int ra[4] = {13, 15, 26, 6};
  const int rb[4] = {17, 29, 16, 24};
  for (int r = 0; r < 4; r++) { x0 += x1; x1 = rotl32(x1, ra[r]); x1 ^= x0; }
  x0 += ks1; x1 += ks2 + 1u;
  for (int r = 0; r < 4; r++) { x0 += x1; x1 = rotl32(x1, rb[r]); x1 ^= x0; }
  x0 += ks2; x1 += ks0 + 2u;
  for (int r = 0; r < 4; r++) { x0 += x1; x1 = rotl32(x1, ra[r]); x1 ^= x0; }
  x0 += ks0; x1 += ks1 + 3u;
  for (int r = 0; r < 4; r++) { x0 += x1; x1 = rotl32(x1, rb[r]); x1 ^= x0; }
  x0 += ks1; x1 += ks2 + 4u;
  for (int r = 0; r < 4; r++) { x0 += x1; x1 = rotl32(x1, ra[r]); x1 ^= x0; }
  x0 += ks2; x1 += ks0 + 5u;
  return tf2{x0, x1};
}

// jax.random.key(42) -> (0,42); split -> kp,kn : folded at compile time
constexpr tf2 SPLIT_A = threefry2x32(0u, 42u, 0u, 2u);
constexpr tf2 SPLIT_B = threefry2x32(0u, 42u, 1u, 3u);
constexpr uint32_t KP0 = SPLIT_A.x0, KP1 = SPLIT_B.x0;
constexpr uint32_t KN0 = SPLIT_A.x1, KN1 = SPLIT_B.x1;

// --------------------------- 1) class lists --------------------------------
// Parallel stable counting-sort. 128 threads, 64 labels each.
// lcnt[c*128 + t] : thread-t count (then exclusive prefix) for class c.
__global__ __launch_bounds__(128) void build_lists(const int* __restrict__ labels,
                                                   int* __restrict__ ws) {
  __shared__ int lcnt[NCLS * 128];            // 32 KB
  const int t = threadIdx.x;
  const int base = t * (B_SZ / 128);

  for (int j = t; j < NCLS * 128; j += 128) lcnt[j] = 0;
  __syncthreads();

#pragma unroll 4
  for (int j = 0; j < B_SZ / 128; ++j) lcnt[labels[base + j] * 128 + t]++;
  __syncthreads();

  if (t < NCLS) {                             // class c = t: prefix over threads
    int run = 0;
    for (int u = 0; u < 128; ++u) { int v = lcnt[t * 128 + u]; lcnt[t * 128 + u] = run; run += v; }
    ws[WS_CNT + t] = run;
  }
  __syncthreads();

  if (t == 0) {
    int acc = 0;
    for (int j = 0; j < NCLS; ++j) { ws[WS_OFF + j] = acc; acc += ws[WS_CNT + j]; }
    ((float*)ws)[WS_ACC + 0] = 0.0f;
    ws[WS_ACC + 1] = 0;
    ws[WS_ACC + 2] = 0;
  }
  __syncthreads();

  for (int j = 0; j < B_SZ / 128; ++j) {      // stable scatter
    const int i = base + j;
    const int c = labels[i];
    const int rank = lcnt[c * 128 + t]++;     // global in-class rank
    ws[WS_LIST + ws[WS_OFF + c] + rank] = i;
    ws[WS_RANK + i] = rank;
  }
}

// ------------------------------ 2) norms -----------------------------------
__global__ void norms_kernel(const float* __restrict__ emb, int* __restrict__ ws) {
  float* nrm = (float*)(ws + WS_NORM);
  const int lane = threadIdx.x & 31;
  const int wave = threadIdx.x >> 5;
  const int row  = blockIdx.x * 8 + wave;
  const float4 v = ((const float4*)(emb + (size_t)row * DIM))[lane];
  float s = v.x * v.x + v.y * v.y + v.z * v.z + v.w * v.w;
#pragma unroll
  for (int o = 16; o > 0; o >>= 1) s += __shfl_xor(s, o, 32);
  if (lane == 0) nrm[row] = s;
}

// ----------------------------- 3) sampling ---------------------------------
__global__ void sample_kernel(const int* __restrict__ labels, int* __restrict__ ws) {
  const int m = blockIdx.x * blockDim.x + threadIdx.x;
  if (m >= B_SZ * K_TRIP) return;
  const int H = (B_SZ * K_TRIP) / 2;
  const int i = m / K_TRIP;
  const int l = labels[i];
  const int cnt_l = ws[WS_CNT + l];
  const int off_l = ws[WS_OFF + l];
  const int* list = ws + WS_LIST;

  // uniform(key,(B,K)): counts iota(B*K), halves paired (m, m+H)
  const uint32_t c0 = (m < H) ? (uint32_t)m : (uint32_t)(m - H);
  const uint32_t c1 = (m < H) ? (uint32_t)(m + H) : (uint32_t)m;
  const tf2 rp2 = threefry2x32(KP0, KP1, c0, c1);
  const tf2 rn2 = threefry2x32(KN0, KN1, c0, c1);
  const uint32_t bp = (m < H) ? rp2.x0 : rp2.x1;
  const uint32_t bn = (m < H) ? rn2.x0 : rn2.x1;
  const float up = __uint_as_float((bp >> 9) | 0x3f800000u) - 1.0f;
  const float un = __uint_as_float((bn >> 9) | 0x3f800000u) - 1.0f;

  const int pos_cnt = cnt_l - 1;
  const int neg_cnt = B_SZ - cnt_l;
  int rp = min((int)(up * (float)pos_cnt), max(pos_cnt - 1, 0)); rp = max(rp, 0);
  int rn = min((int)(un * (float)neg_cnt), max(neg_cnt - 1, 0)); rn = max(rn, 0);

  // positive: rank within class skipping self
  const int self_rank = ws[WS_RANK + i];
  int pr = (rp < self_rank) ? rp : rp + 1;
  pr = min(pr, max(cnt_l - 1, 0));
  const int pidx = (pos_cnt > 0) ? list[off_l + pr] : 0;

  // negative: rn-th non-member = rn + m0, m0 = smallest t with list[t]-t > rn
  int lo = 0, hi = cnt_l;
  while (lo < hi) {
    const int mid = (lo + hi) >> 1;
    if (list[off_l + mid] - mid > rn) hi = mid; else lo = mid + 1;
  }
  int nidx = rn + lo;
  nidx = min(max(nidx, 0), B_SZ - 1);

  ws[WS_POSIDX + m] = pidx;
  ws[WS_NEGIDX + m] = nidx;
}

// --------------------------- 4) WMMA triplet -------------------------------
// 4 waves/block, one 16-anchor tile per wave (private LDS region; same-wave
// ordering only, no barrier). Anchor tile staged with async global->LDS b128
// when the toolchain exposes it (ASYNCcnt-tracked, bypasses VGPRs). Per GEMM:
// B fragments bulk-loaded (clause-grouped), then WMMA bursts on matrix pipe.
__global__ __launch_bounds__(128) void triplet_kernel(const float* __restrict__ emb,
                                                      const int* __restrict__ labels,
                                                      int* __restrict__ ws) {
  __shared__ float atile_all[4][16 * DIM];    // 32 KB, 8 KB per wave
  const int lane = threadIdx.x & 31;
  const int wave = threadIdx.x >> 5;
  const int base = (blockIdx.x * 4 + wave) * 16;
  const int col   = lane & 15;                // N column / anchor within tile
  const int khalf = (lane >> 4) * 2;          // K sub-offset for this half-wave
  float* atile = atile_all[wave];

  // stage this wave's 16 anchor rows (8 KB) into LDS
  {
    const float4* src = (const float4*)(emb + (size_t)base * DIM);
    float4* dst = (float4*)atile;
#if USE_ASYNC_LDS
#pragma unroll
    for (int t = 0; t < 16; ++t)
      __builtin_amdgcn_global_load_async_to_lds_b128(
          (AS_GLOBAL v4i*)(v4i*)(src + lane + 32 * t),
          (AS_LDS    v4i*)(v4i*)(dst + lane + 32 * t), 0, 0);
    __builtin_amdgcn_s_wait_asynccnt(0);
#else
#pragma unroll
    for (int t = 0; t < 16; ++t) dst[lane + 32 * t] = src[lane + 32 * t];
#endif
  }

  // preload all A fragments once (reused by all 10 GEMMs): 32 x v2f
  v2f afrag[32];
#pragma unroll
  for (int s = 0; s < 32; ++s)
    afrag[s] = *(const v2f*)(atile + col * DIM + s * 4 + khalf);

  const int* pidxs = ws + WS_POSIDX;
  const int* nidxs = ws + WS_NEGIDX;
  float dots[10];

#pragma unroll
  for (int g = 0; g < 10; ++g) {
    const int anchor = base + col;
    const int p = (g < 5) ? pidxs[anchor * K_TRIP + g]
                          : nidxs[anchor * K_TRIP + (g - 5)];
    const float* prow = emb + (size_t)p * DIM;
    __builtin_prefetch(prow, 0, 1);           // global_prefetch_b8

    // bulk-load B fragments (gathered b64 clauses), then WMMA burst
    v2f bfrag[32];
#pragma unroll
    for (int s = 0; s < 32; ++s)
      bfrag[s] = *(const v2f*)(prow + s * 4 + khalf);

    v8f acc = {};
#pragma unroll
    for (int s = 0; s < 32; ++s)
      acc = __builtin_amdgcn_wmma_f32_16x16x4_f32(false, afrag[s], false, bfrag[s],
                                                  (short)0, acc, false, false);

    // diag(i): VGPR (i&7), source lane (i<8 ? i : i+16); wave-uniform shuffles
    const int srcl = (col < 8) ? col : col + 16;
    float sh[8];
#pragma unroll
    for (int r = 0; r < 8; ++r) sh[r] = __shfl(acc[r], srcl, 32);
    const int rsel = col & 7;
    float d = sh[0];
#pragma unroll
    for (int r = 1; r < 8; ++r) d = (rsel == r) ? sh[r] : d;
    dots[g] = d;
  }

  // per-anchor loss (lanes 0..15), then full-wave reduction
  const float* nrm = (const float*)(ws + WS_NORM);
  float lsum = 0.0f, lact = 0.0f, lval = 0.0f;
  if (lane < 16) {
    const int anchor = base + lane;
    const int l = labels[anchor];
    const int cnt_l = ws[WS_CNT + l];
    const bool valid = (cnt_l > 1) && (B_SZ - cnt_l > 0);
    const float nx = nrm[anchor];
    const float margin = 1.0f + 2.0f * sqrtf(nx); // MARGIN*(1+BOUNDARY*||a||)
#pragma unroll
    for (int k = 0; k < K_TRIP; ++k) {
      const int pi = pidxs[anchor * K_TRIP + k];
      const int ni = nidxs[anchor * K_TRIP + k];
      const float nyp = nrm[pi], nyn = nrm[ni];
      float sqp = fmaxf(nx + nyp - 2.0f * dots[k], 0.0f);
      float sqn = fmaxf(nx + nyn - 2.0f * dots[5 + k], 0.0f);
      float dep = fmaxf((1.0f - nx) * (1.0f - nyp), EPSF);
      float den = fmaxf((1.0f - nx) * (1.0f - nyn), EPSF);
      float ap = fmaxf(1.0f + 2.0f * sqp / dep, 1.0f + EPSF);
      float an = fmaxf(1.0f + 2.0f * sqn / den, 1.0f + EPSF);
      float dp = logf(ap + sqrtf(ap * ap - 1.0f)); // arccosh
      float dn = logf(an + sqrtf(an * an - 1.0f));
      float tt = fmaxf(dp - dn + margin, 0.0f);
      if (valid) { lsum += tt; lact += (tt > 0.0f) ? 1.0f : 0.0f; }
    }
    lval = valid ? 1.0f : 0.0f;
  }
#pragma unroll
  for (int o = 16; o > 0; o >>= 1) {
    lsum += __shfl_xor(lsum, o, 32);
    lact += __shfl_xor(lact, o, 32);
    lval += __shfl_xor(lval, o, 32);
  }
  if (lane == 0) {
    atomicAdd((float*)ws + WS_ACC + 0, lsum);
    atomicAdd(ws + WS_ACC + 1, (int)lact);
    atomicAdd(ws + WS_ACC + 2, (int)lval);
  }
}

// ------------------------------ 5) finalize --------------------------------
__global__ void finalize_kernel(const int* __restrict__ ws, float* __restrict__ out) {
  const float s   = ((const float*)ws)[WS_ACC + 0];
  const int   act = ws[WS_ACC + 1];
  const int   val = ws[WS_ACC + 2];
  const int   total = val * K_TRIP;
  const float denom = (float)max(total, 1);
  out[0] = s / denom;
  out[1] = (float)act;
  out[2] = (float)total;
  out[3] = (float)act / denom;
}

// ---------------------------------------------------------------------------
extern "C" void kernel_launch(void* const* d_in, const int* in_sizes, int n_in,
                              void* d_out, int out_size, void* d_ws, size_t ws_size,
                              hipStream_t stream) {
  const float* emb    = (const float*)d_in[0];
  const int*   labels = (const int*)d_in[1];
  int*   ws  = (int*)d_ws;
  float* out = (float*)d_out;

  build_lists<<<1, 128, 0, stream>>>(labels, ws);
  norms_kernel<<<B_SZ / 8, 256, 0, stream>>>(emb, ws);
  sample_kernel<<<(B_SZ * K_TRIP + 255) / 256, 256, 0, stream>>>(labels, ws);
  triplet_kernel<<<B_SZ / 64, 128, 0, stream>>>(emb, labels, ws);
  finalize_kernel<<<1, 1, 0, stream>>>(ws, out);
}